// PICSimulation_53197464929166
// MI455X (gfx1250) — compile-verified
//
#include <hip/hip_runtime.h>
#include <hip/hip_bf16.h>
#include <math.h>

// ---- static simulation parameters (mirror reference) ----
#define N_PART   262144
#define NMESH    1024
#define NSTEPS   50
__device__ __host__ constexpr float BOX   = 50.0f;
__device__ __host__ constexpr float DT    = 0.1f;
__device__ __host__ constexpr float DXI   = 1024.0f / 50.0f;                         // 1/DX
__device__ __host__ constexpr float W0    = (50.0f / 262144.0f) * (1024.0f / 50.0f); // = 1/256
__device__ __host__ constexpr float PI_F  = 3.14159265358979f;

typedef float v2f __attribute__((ext_vector_type(2)));
typedef float v8f __attribute__((ext_vector_type(8)));

// floored mod matching jnp.mod for scalars
__device__ __forceinline__ float fmod_pos(float a) {
    return a - floorf(a * (1.0f / BOX)) * BOX;
}

// ---- one-time: circulant Green's function of the FFT Poisson solve ----
// E[n] = sum_m g[(n-m) mod N] * (W0*rho_counts[m]);
// g[d] = -(L/(pi*N)) * sum_{k=1}^{N/2-1} sin(2*pi*k*d/N)/k
__global__ __launch_bounds__(256) void pic_init_g(float* __restrict__ g) {
    int d = blockIdx.x * 256 + threadIdx.x;   // 0..1023
    float s = 0.0f;
    for (int k = 1; k <= NMESH / 2 - 1; ++k) {
        int m = (k * d) & (NMESH - 1);        // exact integer phase reduction
        s += sinf((float)m * (2.0f * PI_F / (float)NMESH)) / (float)k;
    }
    g[d] = -(BOX / (PI_F * (float)NMESH)) * s;
}

__global__ __launch_bounds__(256) void pic_zero(float* __restrict__ rho) {
    int i = blockIdx.x * 256 + threadIdx.x;
    if (i < NMESH) rho[i] = 0.0f;
}

// ---- initial state + CIC deposit of p0 via LDS-privatized histogram ----
__global__ __launch_bounds__(256) void pic_init_deposit(const float* __restrict__ pos_in,
                                                        const float* __restrict__ vel_in,
                                                        float* __restrict__ pos,
                                                        float* __restrict__ vel,
                                                        float* __restrict__ rho) {
    __shared__ float hist[NMESH];
    for (int i = threadIdx.x; i < NMESH; i += 256) hist[i] = 0.0f;
    __syncthreads();
    int base = blockIdx.x * 256 * 8;
    #pragma unroll
    for (int it = 0; it < 8; ++it) {
        int i = base + it * 256 + threadIdx.x;
        float p = fmod_pos(pos_in[i]);
        pos[i] = p;
        vel[i] = vel_in[i];
        float x    = p * DXI;
        float jf   = floorf(x);
        int   j    = ((int)jf) & (NMESH - 1);
        float frac = x - jf;
        atomicAdd(&hist[j], 1.0f - frac);                  // ds_add_f32
        atomicAdd(&hist[(j + 1) & (NMESH - 1)], frac);
    }
    __syncthreads();
    for (int i = threadIdx.x; i < NMESH; i += 256) {
        float w = hist[i];
        if (w != 0.0f) atomicAdd(&rho[i], w);              // global_atomic_add_f32
    }
}

// ---- Poisson solve as dense circulant matvec on the matrix core ----
// E[16p+q] = sum_{d=0}^{63} sum_{s=0}^{15} g[(16d+q-s)&1023] * x[16*((p-d)&63)+s]
// One wave per 16-wide p-group (P = waveId); 64 d-shifts x 4 K-chunks of
// V_WMMA_F32_16X16X4_F32 -> 1024 fully dense f32 WMMAs for the whole 1024-pt grid.
// g and rho are staged DOUBLED in LDS so the inner loop needs no modular masking.
// Also zeroes the ping-pong rho buffer for the next deposit (free: 1 block).
__global__ __launch_bounds__(128) void pic_poisson_wmma(const float* __restrict__ rho_in,
                                                        const float* __restrict__ g,
                                                        float* __restrict__ E,
                                                        float* __restrict__ rho_zero) {
    __shared__ float gsd[1056];   // gsd[i] = g[(i-16) & 1023], i in 0..1055
    __shared__ float xsd[2048];   // doubled W0-scaled rho (column wrap without masks)
    int t = threadIdx.x;
    for (int i = t; i < NMESH; i += 128) {
        float x = rho_in[i] * W0;         // CIC normalization folded in
        xsd[i]          = x;
        xsd[i + NMESH]  = x;
        rho_zero[i]     = 0.0f;           // prep next step's deposit target
    }
    for (int i = t; i < 1056; i += 128) gsd[i] = g[(i + NMESH - 16) & (NMESH - 1)];
    __syncthreads();

    int P  = t >> 5;        // wave id = p-group 0..3
    int l  = t & 31;        // lane
    int lh = l >> 4;        // lane half (ISA VGPR layout)
    int ll = l & 15;

    v8f c = {};             // 16x16 f32 accumulator (8 VGPRs)
    for (int d = 0; d < 64; ++d) {
        int cb = (16 * P + ll + 64 - d) * 16;   // B column base (in [16,2032])
        int ab = 16 * d + ll + 16;              // A row base   (in [0,1039])
        #pragma unroll
        for (int cc = 0; cc < 4; ++cc) {
            int kA = 4 * cc + 2 * lh;           // K = reg + 2*lanehalf (ISA A/B layout)
            v2f a, b;
            a.y = gsd[ab - kA - 1];
            a.x = gsd[ab - kA];
            b.x = xsd[cb + kA];
            b.y = xsd[cb + kA + 1];
            // 8 args: (neg_a, A, neg_b, B, c_mod, C, reuse_a, reuse_b)
            c = __builtin_amdgcn_wmma_f32_16x16x4_f32(false, a, false, b,
                                                      (short)0, c, false, false);
        }
    }
    // D layout: reg v, lane l -> M = v + 8*lh (=q), N = ll (=pp); E[16*(16P+pp)+q]
    int base = 16 * (16 * P + ll) + 8 * lh;
    #pragma unroll
    for (int v = 0; v < 8; ++v) E[base + v] = c[v];
}

// ---- fused step kernel ----
// Finishes step n-1: a = CIC-gather(E) at current positions, v += a*dt/2, emit traj.
// Starts step n: half-kick, drift, CIC deposit into the ping-pong rho buffer.
// out_prev == nullptr -> first call (a is the initial acceleration; no traj emit).
// rho_next == nullptr -> final call (emit only; no further push/deposit).
__global__ __launch_bounds__(256) void pic_update(float* __restrict__ pos,
                                                  float* __restrict__ vel,
                                                  const float* __restrict__ E,
                                                  float* __restrict__ rho_next,
                                                  float* __restrict__ out_prev) {
    __shared__ float Es[NMESH];
    __shared__ float hist[NMESH];
    for (int i = threadIdx.x; i < NMESH; i += 256) { Es[i] = E[i]; hist[i] = 0.0f; }
    __syncthreads();

    int base = blockIdx.x * 256 * 8;
    #pragma unroll
    for (int it = 0; it < 8; ++it) {
        int i = base + it * 256 + threadIdx.x;
        float p    = pos[i];
        float x    = p * DXI;
        float jf   = floorf(x);
        int   j    = ((int)jf) & (NMESH - 1);
        float frac = x - jf;
        float a    = -((1.0f - frac) * Es[j] + frac * Es[(j + 1) & (NMESH - 1)]); // -Q*E/M
        float v    = vel[i];
        if (out_prev) {
            v += a * (DT * 0.5f);                       // finish previous step's kick
            float* o = out_prev + (size_t)i * 3;        // (pos, vel, acc) -> b96 store
            o[0] = p; o[1] = v; o[2] = a;
        }
        if (rho_next) {
            float vh = v + a * (DT * 0.5f);             // next step: half kick
            float pn = fmod_pos(p + vh * DT);           // drift
            pos[i] = pn;
            vel[i] = vh;
            float xn = pn * DXI;
            float jfn = floorf(xn);
            int   jn  = ((int)jfn) & (NMESH - 1);
            float fn  = xn - jfn;
            atomicAdd(&hist[jn], 1.0f - fn);
            atomicAdd(&hist[(jn + 1) & (NMESH - 1)], fn);
        }
    }
    __syncthreads();
    if (rho_next) {
        for (int i = threadIdx.x; i < NMESH; i += 256) {
            float w = hist[i];
            if (w != 0.0f) atomicAdd(&rho_next[i], w);
        }
    }
}

extern "C" void kernel_launch(void* const* d_in, const int* in_sizes, int n_in,
                              void* d_out, int out_size, void* d_ws, size_t ws_size,
                              hipStream_t stream) {
    const float* pos_in = (const float*)d_in[0];
    const float* vel_in = (const float*)d_in[1];
    float* out = (float*)d_out;

    float* ws      = (float*)d_ws;
    float* cur_pos = ws;
    float* cur_vel = ws + (size_t)N_PART;
    float* rhoA    = ws + (size_t)2 * N_PART;
    float* rhoB    = rhoA + NMESH;
    float* Eg      = rhoB + NMESH;
    float* gk      = Eg + NMESH;
    float* rho[2]  = { rhoA, rhoB };

    const int DBLK = 128;   // 128 blocks x 256 threads x 8 particles

    // one-time setup (recomputed each launch: deterministic, no cached state)
    pic_init_g<<<NMESH / 256, 256, 0, stream>>>(gk);
    pic_zero<<<NMESH / 256, 256, 0, stream>>>(rhoA);
    pic_init_deposit<<<DBLK, 256, 0, stream>>>(pos_in, vel_in, cur_pos, cur_vel, rhoA);

    for (int n = 0; n < NSTEPS; ++n) {
        pic_poisson_wmma<<<1, 128, 0, stream>>>(rho[n & 1], gk, Eg, rho[(n + 1) & 1]);
        pic_update<<<DBLK, 256, 0, stream>>>(cur_pos, cur_vel, Eg, rho[(n + 1) & 1],
                                             n ? out + (size_t)(n - 1) * N_PART * 3
                                               : nullptr);
    }
    // finish step 49: solve on final deposit, gather, emit last trajectory slice
    pic_poisson_wmma<<<1, 128, 0, stream>>>(rho[NSTEPS & 1], gk, Eg, rho[(NSTEPS + 1) & 1]);
    pic_update<<<DBLK, 256, 0, stream>>>(cur_pos, cur_vel, Eg, nullptr,
                                         out + (size_t)(NSTEPS - 1) * N_PART * 3);
}